// H2GCNConv_33217277067915
// MI455X (gfx1250) — compile-verified
//
#include <hip/hip_runtime.h>

// H2GCNConv: out[:, :128] = A1 @ x ; out[:, 128:] = A2 @ x
// N = 100000, D = 128. Memory-bound scatter SpMM on MI455X:
//  - wave-per-edge (wave32): lane l owns features [4l,4l+4) -> 512B coalesced row gather
//    (x = 51MB, resident in 192MB L2)
//  - edge stream staged into per-wave LDS double buffers via CDNA5 async-to-LDS DMA:
//    3 async ops (2x b128 + 1x b64) per 64-edge chunk, pipelined with s_wait_asynccnt
//  - non-returning global_atomic_add_f32 scatter (resolved in L2 atomic units)
//  - global_prefetch_b8 pipelines upcoming gathered rows

#if __has_builtin(__builtin_amdgcn_global_load_async_to_lds_b128) && \
    __has_builtin(__builtin_amdgcn_global_load_async_to_lds_b64) && \
    __has_builtin(__builtin_amdgcn_s_wait_asynccnt)
#define USE_ASYNC_LDS 1
#else
#define USE_ASYNC_LDS 0
#endif

typedef unsigned long long u64;
typedef unsigned int       u32;
typedef int v2i __attribute__((vector_size(8)));    // 64-bit async payload
typedef int v4i __attribute__((vector_size(16)));   // 128-bit async payload

#define AS1 __attribute__((address_space(1)))
#define AS3 __attribute__((address_space(3)))

static constexpr int TPB            = 256;   // 8 waves per block (wave32)
static constexpr int WPB            = TPB / 32;
static constexpr int SCATTER_BLOCKS = 2048;  // 16384 waves in flight
static constexpr int CHUNK          = 64;    // edges per wave chunk (pairs per lane)

__global__ __launch_bounds__(TPB)
void zero_f4_kernel(float4* __restrict__ out, int n4) {
    int i = blockIdx.x * blockDim.x + threadIdx.x;
    float4 z; z.x = 0.f; z.y = 0.f; z.z = 0.f; z.w = 0.f;
    if (i < n4) out[i] = z;
}

#if USE_ASYNC_LDS
__device__ __forceinline__ void async_g2l_b128(const void* g, void* l) {
    void* gnc = const_cast<void*>(g);
    __builtin_amdgcn_global_load_async_to_lds_b128(
        (AS1 v4i*)gnc, (AS3 v4i*)l, /*offset=*/0, /*cpol=*/0);
}
__device__ __forceinline__ void async_g2l_b64(const void* g, void* l) {
    void* gnc = const_cast<void*>(g);
    __builtin_amdgcn_global_load_async_to_lds_b64(
        (AS1 v2i*)gnc, (AS3 v2i*)l, /*offset=*/0, /*cpol=*/0);
}
#endif

__global__ __launch_bounds__(TPB)
void spmm_scatter_kernel(const long long* __restrict__ dst,   // edge_index[0]
                         const long long* __restrict__ src,   // edge_index[1]
                         const float*     __restrict__ val,
                         const float4*    __restrict__ x4,    // [N,32] float4 view of [N,128]
                         float*           __restrict__ out,   // [N,256]
                         int E, int coloff) {
    __shared__ __align__(16) u64   ls_src[2][WPB][CHUNK];   // 16KB
    __shared__ __align__(16) u64   ls_dst[2][WPB][CHUNK];
    __shared__ __align__(16) float ls_val[2][WPB][CHUNK];   //  4KB

    const int lane = threadIdx.x & 31;
    const int w    = threadIdx.x >> 5;
    const int gw   = blockIdx.x * WPB + w;
    const int ngw  = gridDim.x * WPB;

    const int nfull = E / CHUNK;   // full 64-edge chunks through the async pipeline

    // ---- stage one full chunk into this wave's LDS slice: 3 async DMA ops ----
    auto stage = [&](int kk, int buf) {
        const int ei = kk * CHUNK + 2 * lane;   // this lane's pair (16B aligned)
#if USE_ASYNC_LDS
        async_g2l_b128(src + ei, &ls_src[buf][w][2 * lane]);
        async_g2l_b128(dst + ei, &ls_dst[buf][w][2 * lane]);
        async_g2l_b64 (val + ei, &ls_val[buf][w][2 * lane]);
#else
        ls_src[buf][w][2 * lane]     = (u64)__builtin_nontemporal_load(&src[ei]);
        ls_src[buf][w][2 * lane + 1] = (u64)__builtin_nontemporal_load(&src[ei + 1]);
        ls_dst[buf][w][2 * lane]     = (u64)__builtin_nontemporal_load(&dst[ei]);
        ls_dst[buf][w][2 * lane + 1] = (u64)__builtin_nontemporal_load(&dst[ei + 1]);
        ls_val[buf][w][2 * lane]     =      __builtin_nontemporal_load(&val[ei]);
        ls_val[buf][w][2 * lane + 1] =      __builtin_nontemporal_load(&val[ei + 1]);
#endif
    };

    if (gw < nfull) {
        int k   = gw;
        int cur = 0;
        stage(k, cur);                           // prime pipeline (ASYNCcnt = 3)

        while (true) {
            const int  kn   = k + ngw;
            const bool more = (kn < nfull);
            if (more) {
                stage(kn, cur ^ 1);              // ASYNCcnt -> 6
#if USE_ASYNC_LDS
                __builtin_amdgcn_s_wait_asynccnt(3);  // chunk k landed (in-order retire)
#endif
            } else {
#if USE_ASYNC_LDS
                __builtin_amdgcn_s_wait_asynccnt(0);
#endif
            }

            // ---- process 64 edges from LDS (broadcast ds reads, low dword of i64) ----
            const u32*   s32 = (const u32*)ls_src[cur][w];
            const u32*   d32 = (const u32*)ls_dst[cur][w];
            const float* vf  = ls_val[cur][w];

            #pragma unroll 4
            for (int i = 0; i < CHUNK; ++i) {
                const int   s = (int)s32[2 * i];
                const int   d = (int)d32[2 * i];
                const float v = vf[i];

                // branchless lookahead prefetch of an upcoming row's 16B lane chunk
                const int ip = (i + 3 < CHUNK) ? (i + 3) : (CHUNK - 1);
                const int sp = (int)s32[2 * ip];
                __builtin_prefetch((const void*)(x4 + (size_t)sp * 32 + lane), 0, 1);

                // coalesced 512B row gather: 32 lanes x global_load_b128 (L2-resident x)
                const float4 xv = x4[(size_t)s * 32 + lane];

                // non-returning global_atomic_add_f32 into out[d, coloff + 4*lane ..]
                float* o = out + (size_t)d * 256 + coloff + lane * 4;
                unsafeAtomicAdd(o + 0, v * xv.x);
                unsafeAtomicAdd(o + 1, v * xv.y);
                unsafeAtomicAdd(o + 2, v * xv.z);
                unsafeAtomicAdd(o + 3, v * xv.w);
            }

            if (!more) break;
            k = kn;
            cur ^= 1;
        }
    }

    // ---- tail edges (E % 64; zero for this problem's sizes) by global wave 0 ----
    if (gw == 0) {
        for (int e = nfull * CHUNK; e < E; ++e) {
            const int   s = (int)src[e];
            const int   d = (int)dst[e];
            const float v = val[e];
            const float4 xv = x4[(size_t)s * 32 + lane];
            float* o = out + (size_t)d * 256 + coloff + lane * 4;
            unsafeAtomicAdd(o + 0, v * xv.x);
            unsafeAtomicAdd(o + 1, v * xv.y);
            unsafeAtomicAdd(o + 2, v * xv.z);
            unsafeAtomicAdd(o + 3, v * xv.w);
        }
    }
}

extern "C" void kernel_launch(void* const* d_in, const int* in_sizes, int n_in,
                              void* d_out, int out_size, void* d_ws, size_t ws_size,
                              hipStream_t stream) {
    const float*     x   = (const float*)d_in[0];
    const long long* ei1 = (const long long*)d_in[1];  // int64 [2, E1]
    const float*     ev1 = (const float*)d_in[2];
    const long long* ei2 = (const long long*)d_in[3];  // int64 [2, E2]
    const float*     ev2 = (const float*)d_in[4];
    float*           out = (float*)d_out;

    const int E1 = in_sizes[2];
    const int E2 = in_sizes[4];

    // 1) zero the accumulator (harness poisons d_out; must re-zero every call)
    const int n4 = out_size / 4;
    zero_f4_kernel<<<(n4 + TPB - 1) / TPB, TPB, 0, stream>>>((float4*)out, n4);

    // 2) graph 1 -> columns [0,128)
    spmm_scatter_kernel<<<SCATTER_BLOCKS, TPB, 0, stream>>>(
        ei1, ei1 + E1, ev1, (const float4*)x, out, E1, /*coloff=*/0);

    // 3) graph 2 -> columns [128,256)
    spmm_scatter_kernel<<<SCATTER_BLOCKS, TPB, 0, stream>>>(
        ei2, ei2 + E2, ev2, (const float4*)x, out, E2, /*coloff=*/128);
}